// TimeWarpingLayer_31396210934232
// MI455X (gfx1250) — compile-verified
//
#include <hip/hip_runtime.h>
#include <math.h>

// ---------------------------------------------------------------------------
// TimeWarpingLayer for MI455X (gfx1250, wave32).
//
// Roofline: 128 MiB in + 128 MiB out @ 23.3 TB/s => ~11 us floor. The 25-tap
// Gaussian FIR (50 FLOP/elem, 1.7 GFLOP fp32) + warp lerp are fused into ONE
// banded matmul per 16x16 output tile on V_WMMA_F32_16X16X4_F32, so the FIR
// rides the matrix pipe instead of VALU. The 192 MB L2 holds the whole 128 MiB
// input, so the ~2.6x halo re-read between adjacent s-tiles hits L2.
//
//   out[s0+i, f] = sum_r coeff_i(r) * x[r, f]
//   coeff_i(r)   = (1-frac_i)*kern[r-i0_i+12] + frac_i*kern[r-i0_i+11]
// and since i0 is monotone (softplus > 0), rows r sweep one contiguous band
// [i0[s0]-12, i0[s0+15]+13], 4 K-rows per WMMA step (data-dependent count).
// ---------------------------------------------------------------------------

#define SLEN   2048
#define FEAT   512
#define BATCH  32
#define RAD    12
#define KSIZE  25   // 2*RAD+1
#define KPAD   3    // zero padding on each flank of the LDS tap table

typedef float v2f __attribute__((ext_vector_type(2)));
typedef float v8f __attribute__((ext_vector_type(8)));

// ---------------------------------------------------------------------------
// Kernel A: warp computation (single workgroup).
// ---------------------------------------------------------------------------
__global__ __launch_bounds__(256)
void tw_warp_kernel(const float* __restrict__ W1, const float* __restrict__ b1,
                    const float* __restrict__ W2, const float* __restrict__ b2,
                    int* __restrict__ i0_out, float* __restrict__ frac_out,
                    float* __restrict__ loss_out)
{
    __shared__ float lds_sum[256];
    __shared__ float warp_s[SLEN];
    const int tid = threadIdx.x;
    const float b2v = b2[0];

    // Each thread owns 8 consecutive time steps.
    float incl[8];
    float run = 0.f;
    #pragma unroll
    for (int j = 0; j < 8; ++j) {
        const int s = tid * 8 + j;
        const float t = (float)s / (float)(SLEN - 1);
        float acc = b2v;
        for (int k = 0; k < 64; ++k) {            // W1:[64,1], W2:[1,64]
            float h = fmaf(t, W1[k], b1[k]);
            h = fmaxf(h, 0.f);                    // relu
            acc = fmaf(W2[k], h, acc);
        }
        const float w = tanhf(acc);
        // softplus(w) = max(w,0) + log1p(exp(-|w|))
        const float sp = fmaxf(w, 0.f) + log1pf(expf(-fabsf(w)));
        run += sp;
        incl[j] = run;                            // inclusive within chunk
    }
    lds_sum[tid] = run;
    __syncthreads();
    // Hillis–Steele inclusive scan of 256 chunk totals.
    for (int off = 1; off < 256; off <<= 1) {
        const float add = (tid >= off) ? lds_sum[tid - off] : 0.f;
        __syncthreads();
        lds_sum[tid] += add;
        __syncthreads();
    }
    const float excl  = (tid > 0) ? lds_sum[tid - 1] : 0.f;
    const float total = lds_sum[255];
    const float inv_total = 1.f / total;

    #pragma unroll
    for (int j = 0; j < 8; ++j) {
        const int s = tid * 8 + j;
        const float wn = (excl + incl[j]) * inv_total;   // warp1[s]
        warp_s[s] = wn;
        const float pos = wn * (float)(SLEN - 1);
        int i0 = (int)floorf(pos);
        i0 = min(max(i0, 0), SLEN - 2);
        i0_out[s]   = i0;
        frac_out[s] = pos - (float)i0;
    }
    __syncthreads();

    // Smoothness loss: mean over (S-2) of (warp[s+2]-2 warp[s+1]+warp[s])^2.
    float acc2 = 0.f;
    #pragma unroll
    for (int j = 0; j < 8; ++j) {
        const int s = tid * 8 + j;
        if (s <= SLEN - 3) {
            const float d2 = warp_s[s + 2] - 2.f * warp_s[s + 1] + warp_s[s];
            acc2 = fmaf(d2, d2, acc2);
        }
    }
    lds_sum[tid] = acc2;
    __syncthreads();
    for (int off = 128; off > 0; off >>= 1) {
        if (tid < off) lds_sum[tid] += lds_sum[tid + off];
        __syncthreads();
    }
    if (tid == 0) loss_out[0] = lds_sum[0] / (float)(SLEN - 2);
}

// 'symmetric' reflection (scipy gaussian_filter1d mode='reflect').
__device__ __forceinline__ int tw_reflect(int r) {
    r = (r < 0) ? (-1 - r) : r;
    r = (r >= SLEN) ? (2 * SLEN - 1 - r) : r;
    return r;
}

// ---------------------------------------------------------------------------
// Kernel B: fused Gaussian-smooth + warp-interp as a banded WMMA matmul.
// One wave per 16(s_out) x 16(feature) tile; 8 waves/block tile 128 features.
// WMMA f32 lane layout (ISA 7.12.2): A: M=L&15, K=2*(L>>4)+v;
//                                    B: N=L&15, K=2*(L>>4)+v;
//                                    C: N=L&15, M=v+8*(L>>4).
// ---------------------------------------------------------------------------
__global__ __launch_bounds__(256)
void tw_smooth_warp_kernel(const float* __restrict__ x,
                           const int*   __restrict__ i0_arr,
                           const float* __restrict__ frac_arr,
                           float* __restrict__ out)
{
    // Zero-padded Gaussian tap table in LDS: lkp[KPAD+t] = kern[t], t=0..24,
    // zero flanks so out-of-band lookups are branchless ds_loads of 0.
    __shared__ float  lkp[KSIZE + 2 * KPAD + 1];   // 32 entries
    __shared__ double dk[KSIZE];
    const int tid = threadIdx.x;
    if (tid < KSIZE + 2 * KPAD + 1) lkp[tid] = 0.f;
    if (tid < KSIZE) {
        const double d = (double)(tid - RAD) / 3.0;   // SIGMA = 3
        dk[tid] = exp(-0.5 * d * d);
    }
    __syncthreads();
    if (tid < KSIZE) {
        double ssum = 0.0;
        for (int i = 0; i < KSIZE; ++i) ssum += dk[i];
        lkp[tid + KPAD] = (float)(dk[tid] / ssum);    // fp64-normalized taps
    }
    __syncthreads();

    const int lane = tid & 31;
    const int wave = tid >> 5;
    const int m    = lane & 15;
    const int half = lane >> 4;
    const int s0 = blockIdx.x * 16;
    const int f0 = (blockIdx.y * 8 + wave) * 16;
    const int b  = blockIdx.z;
    const float* __restrict__ xb = x + (size_t)b * (SLEN * FEAT);
    const int fcol = f0 + m;

    // Per-output-row warp params (lanes 16-31 mirror 0-15).
    const int   i0_l = i0_arr[s0 + m];
    const float fr_l = frac_arr[s0 + m];
    const float omf  = 1.f - fr_l;

    // Band extent for this tile (i0 monotone nondecreasing).
    const int r_lo  = __shfl(i0_l, 0)  - RAD;
    const int r_hi  = __shfl(i0_l, 15) + RAD + 1;
    const int steps = (r_hi - r_lo + 4) >> 2;       // ceil(nrows/4)

    // Per-lane loop invariants. K-row of fragment element v is
    // rbase+v, rbase = r_lo + 4c + 2*half; tap index t0 = rbase - i0 + RAD.
    const int t_base = r_lo + 2 * half - i0_l + RAD;

    v8f acc = {};   // C/D: 16x16 fp32 accumulator

    // Max row touched by the loop is r_lo + 4*steps - 1 <= r_hi + 3.
    if (r_lo >= 0 && r_hi + 3 < SLEN) {
        // ---- interior fast path: no reflection, 32-bit induction offsets ----
        int ib = (r_lo + 2 * half) * FEAT + fcol;   // element offset, >= 0
        int t0 = t_base;
        for (int c = 0; c < steps; ++c) {
            // A fragment: blended taps via padded-table lookups (branchless).
            const int tc = min(max(t0, -2), KSIZE + 1);   // med3(t0,-2,26)
            const float k_m1 = lkp[tc + KPAD - 1];
            const float k_0  = lkp[tc + KPAD];
            const float k_p1 = lkp[tc + KPAD + 1];
            v2f a;
            a.x = omf * k_0  + fr_l * k_m1;   // coeff for row rbase
            a.y = omf * k_p1 + fr_l * k_0;    // coeff for row rbase+1
            // B fragment: rows rbase, rbase+1 (second folds to imm offset).
            v2f bf;
            bf.x = xb[ib];
            bf.y = xb[ib + FEAT];
            acc = __builtin_amdgcn_wmma_f32_16x16x4_f32(
                false, a, false, bf, (short)0, acc, false, false);
            ib += 4 * FEAT;
            t0 += 4;
        }
    } else {
        // ---- edge path: reflect-padded rows (few tiles) ----
        for (int c = 0; c < steps; ++c) {
            const int rbase = r_lo + 4 * c + 2 * half;
            const int t0 = t_base + 4 * c;
            const int tc = min(max(t0, -2), KSIZE + 1);
            const float k_m1 = lkp[tc + KPAD - 1];
            const float k_0  = lkp[tc + KPAD];
            const float k_p1 = lkp[tc + KPAD + 1];
            v2f a;
            a.x = omf * k_0  + fr_l * k_m1;
            a.y = omf * k_p1 + fr_l * k_0;
            v2f bf;
            bf.x = xb[tw_reflect(rbase)     * FEAT + fcol];
            bf.y = xb[tw_reflect(rbase + 1) * FEAT + fcol];
            acc = __builtin_amdgcn_wmma_f32_16x16x4_f32(
                false, a, false, bf, (short)0, acc, false, false);
        }
    }

    // Store C/D: lane L, VGPR v -> (row = s0 + v + 8*half, col = fcol).
    float* __restrict__ ob = out + (size_t)b * (SLEN * FEAT);
    const int obase = (s0 + 8 * half) * FEAT + fcol;
    #pragma unroll
    for (int v = 0; v < 8; ++v) {
        ob[obase + v * FEAT] = acc[v];
    }
}

// ---------------------------------------------------------------------------
extern "C" void kernel_launch(void* const* d_in, const int* in_sizes, int n_in,
                              void* d_out, int out_size, void* d_ws, size_t ws_size,
                              hipStream_t stream)
{
    const float* x  = (const float*)d_in[0];   // [32,2048,512]
    const float* W1 = (const float*)d_in[1];   // [64,1]
    const float* b1 = (const float*)d_in[2];   // [64]
    const float* W2 = (const float*)d_in[3];   // [1,64]
    const float* b2 = (const float*)d_in[4];   // [1]

    float* out  = (float*)d_out;                         // warped, then loss
    float* loss = out + (size_t)BATCH * SLEN * FEAT;     // d_out[B*S*F]

    int*   ws_i0   = (int*)d_ws;                                 // S ints
    float* ws_frac = (float*)((char*)d_ws + SLEN * sizeof(int)); // S floats

    tw_warp_kernel<<<1, 256, 0, stream>>>(W1, b1, W2, b2, ws_i0, ws_frac, loss);

    dim3 grid(SLEN / 16, FEAT / (16 * 8), BATCH);   // (128, 4, 32), 8 waves/blk
    tw_smooth_warp_kernel<<<grid, 256, 0, stream>>>(x, ws_i0, ws_frac, out);
}